// KalmanFilter_14465449853052
// MI455X (gfx1250) — compile-verified
//
#include <hip/hip_runtime.h>
#include <hip/hip_bf16.h>
#include <math.h>

// Kalman filter forward scan, MI455X (gfx1250, wave32).
// B=8192 groups, M=4 meas, N=8 states, T=512 steps.
// 2 groups per wave; P' = F P F^T + Q via 4x V_WMMA_F32_16X16X4_F32 per step;
// measurement update decorrelated (R -> I) into 4 exact sequential scalar updates
// (Joseph form collapses to P -= g g^T / s). All cross-lane traffic is
// wave-private LDS (in-order per wave -> no barriers in the T loop).

#define KF_B 8192
#define KF_M 4
#define KF_N 8
#define KF_T 512

typedef __attribute__((ext_vector_type(2))) float v2f;
typedef __attribute__((ext_vector_type(8))) float v8f;

// ---------------- precompute: Linv = chol(R)^-1, Hp = Linv * H ----------------
__global__ void kf_precompute(const float* __restrict__ R,
                              const float* __restrict__ H,
                              float* __restrict__ ws) {
  if (threadIdx.x != 0 || blockIdx.x != 0) return;
  float L[4][4];
  for (int i = 0; i < 4; i++)
    for (int j = 0; j < 4; j++) L[i][j] = 0.f;
  for (int i = 0; i < 4; i++) {
    for (int j = 0; j <= i; j++) {
      float sum = R[i * 4 + j];
      for (int k = 0; k < j; k++) sum -= L[i][k] * L[j][k];
      L[i][j] = (i == j) ? sqrtf(sum) : sum / L[j][j];
    }
  }
  float Li[4][4];
  for (int i = 0; i < 4; i++)
    for (int j = 0; j < 4; j++) Li[i][j] = 0.f;
  for (int c = 0; c < 4; c++) {              // solve L * X = I (forward subst)
    for (int i = c; i < 4; i++) {
      float sum = (i == c) ? 1.f : 0.f;
      for (int k = 0; k < i; k++) sum -= L[i][k] * Li[k][c];
      Li[i][c] = sum / L[i][i];
    }
  }
  for (int i = 0; i < 16; i++) ws[i] = Li[i / 4][i % 4];      // ws[0..15]  = Linv
  for (int i = 0; i < 4; i++)                                  // ws[16..47] = Hp = Linv*H
    for (int c = 0; c < 8; c++) {
      float sum = 0.f;
      for (int j = 0; j < 4; j++) sum += Li[i][j] * H[j * 8 + c];
      ws[16 + i * 8 + c] = sum;
    }
}

// ---------------- main scan kernel: 8 waves/block, 2 groups/wave ----------------
__global__ __launch_bounds__(256) void kf_kernel(
    const float* __restrict__ x, const float* __restrict__ F,
    const float* __restrict__ H, const float* __restrict__ Q,
    const float* __restrict__ std0, const float* __restrict__ ws,
    float* __restrict__ out) {
  __shared__ float sF[64], sH[32], sHp[32], sLi[16];
  __shared__ __align__(16) float sP[8][2][64];   // covariance, wave-private
  __shared__ __align__(16) float sY[8][2][64];   // Y = P F^T scratch
  __shared__ __align__(16) float sMu[8][2][8];
  __shared__ __align__(16) float sG[8][2][8];    // g = P h^T
  __shared__ float sZ[8][2][4];    // raw obs
  __shared__ float sZp[8][2][4];   // decorrelated obs
  __shared__ float sSc[8][2][2];   // [1/s, innov]

  const int tid = threadIdx.x;
  const int lane = tid & 31;
  const int w = tid >> 5;
  const int half = lane >> 4;      // group within wave (0/1)
  const int col = lane & 15;
  const int g0 = blockIdx.x * 16 + w * 2;   // global groups g0, g0+1
  const int gstr = KF_M * KF_T;

  if (tid < 64) sF[tid] = F[tid];
  if (tid < 32) { sH[tid] = H[tid]; sHp[tid] = ws[16 + tid]; }
  if (tid < 16) sLi[tid] = ws[tid];
  __syncthreads();

  // Loop-invariant WMMA fragments (gfx1250 documented layouts).
  // A 16x4 fp32: VGPR v, lane l -> (M = l%16, K = 4*chunk + v + 2*(l/16))
  // B  4x16 fp32: VGPR v, lane l -> (K = 4*chunk + v + 2*(l/16), N = l%16)
  // C/D 16x16:    VGPR v, lane l -> (M = v + 8*(l/16), N = l%16)
  const int kb = 2 * half;
  v2f aF0, aF1, bFT0, bFT1;
  {
    const int m = col;  // also N for the B fragment
    aF0.x = (m < 8) ? F[m * 8 + (0 + kb)] : 0.f;   // A = [F ; 0]  (16x8)
    aF0.y = (m < 8) ? F[m * 8 + (1 + kb)] : 0.f;
    aF1.x = (m < 8) ? F[m * 8 + (4 + kb)] : 0.f;
    aF1.y = (m < 8) ? F[m * 8 + (5 + kb)] : 0.f;
    bFT0.x = (m < 8) ? F[m * 8 + (0 + kb)] : 0.f;  // B = F^T padded (8x16), B[k][n]=F[n][k]
    bFT0.y = (m < 8) ? F[m * 8 + (1 + kb)] : 0.f;
    bFT1.x = (m < 8) ? F[m * 8 + (4 + kb)] : 0.f;
    bFT1.y = (m < 8) ? F[m * 8 + (5 + kb)] : 0.f;
  }
  v8f cQ;                                    // C = [Q | Q] horizontal (rows 8..15 zero)
#pragma unroll
  for (int v = 0; v < 8; v++) {
    const int m = v + 8 * half;
    cQ[v] = (m < 8) ? Q[m * 8 + (col & 7)] : 0.f;
  }

  // init: P = diag(std^2), mu = 0
#pragma unroll
  for (int q = 0; q < 4; q++) {
    const int idx = lane + 32 * q;           // 0..127 over 2 groups x 64 entries
    const int gr = idx >> 6, rc = idx & 63, r = rc >> 3, c = rc & 7;
    sP[w][gr][rc] = (r == c) ? std0[r] * std0[r] : 0.f;
  }
  if (lane < 16) sMu[w][lane >> 3][lane & 7] = 0.f;

  for (int t = 0; t < KF_T; t++) {
    // (1) load obs + decorrelate z' = Linv z (single EXEC region; wave in-order LDS)
    if (lane < 8) {
      const int g = lane >> 2, m = lane & 3;
      sZ[w][g][m] = x[(g0 + g) * gstr + m * KF_T + t];
      float zp = 0.f;
      for (int j = 0; j <= m; j++) zp += sLi[m * 4 + j] * sZ[w][g][j];
      sZp[w][g][m] = zp;
    }
    // (2) mu <- F mu (in place; all LDS reads precede the single store)
    if (lane < 16) {
      const int g = lane >> 3, r = lane & 7;
      float mp = 0.f;
      for (int c = 0; c < 8; c++) mp += sF[r * 8 + c] * sMu[w][g][c];
      sMu[w][g][r] = mp;
    }
    // (3) out = H mu_pred (original measurement space)
    if (lane < 8) {
      const int g = lane >> 2, m = lane & 3;
      float o = 0.f;
      for (int c = 0; c < 8; c++) o += sH[m * 8 + c] * sMu[w][g][c];
      out[(g0 + g) * gstr + m * KF_T + t] = o;
    }

    // (5) P <- F P F^T + Q  via 4x v_wmma_f32_16x16x4_f32
    v2f a10, a11;  // A = [P0 ; P1] vertical stack (16x8)
    {
      const int gA = col >> 3, rA = col & 7;
      a10.x = sP[w][gA][rA * 8 + (0 + kb)];
      a10.y = sP[w][gA][rA * 8 + (1 + kb)];
      a11.x = sP[w][gA][rA * 8 + (4 + kb)];
      a11.y = sP[w][gA][rA * 8 + (5 + kb)];
    }
    v8f Dy = {};
    Dy = __builtin_amdgcn_wmma_f32_16x16x4_f32(false, a10, false, bFT0, (short)0, Dy, false, false);
    Dy = __builtin_amdgcn_wmma_f32_16x16x4_f32(false, a11, false, bFT1, (short)0, Dy, false, false);
    if (col < 8) {                           // store Y_g (rows v, group=half)
#pragma unroll
      for (int v = 0; v < 8; v++) sY[w][half][v * 8 + col] = Dy[v];
    }
    v2f b20, b21;                            // B = [Y0 | Y1] horizontal (8x16)
    {
      const int gB = col >> 3, cB = col & 7;
      b20.x = sY[w][gB][(0 + kb) * 8 + cB];
      b20.y = sY[w][gB][(1 + kb) * 8 + cB];
      b21.x = sY[w][gB][(4 + kb) * 8 + cB];
      b21.y = sY[w][gB][(5 + kb) * 8 + cB];
    }
    v8f D2 = cQ;
    D2 = __builtin_amdgcn_wmma_f32_16x16x4_f32(false, aF0, false, b20, (short)0, D2, false, false);
    D2 = __builtin_amdgcn_wmma_f32_16x16x4_f32(false, aF1, false, b21, (short)0, D2, false, false);
    if (half == 0) {                         // valid rows 0..7 -> [P'0 | P'1]
      const int gD = col >> 3, cD = col & 7;
#pragma unroll
      for (int v = 0; v < 8; v++) sP[w][gD][v * 8 + cD] = D2[v];
    }

    // (6) four exact sequential scalar measurement updates (R' = I)
#pragma unroll
    for (int i = 0; i < 4; i++) {
      const float* h = &sHp[i * 8];
      if (lane < 16) {                       // g = P h^T (vector LDS loads)
        const int gr = lane >> 3, r = lane & 7;
        const float4 p0 = *(const float4*)&sP[w][gr][r * 8];
        const float4 p1 = *(const float4*)&sP[w][gr][r * 8 + 4];
        float gv = p0.x * h[0] + p0.y * h[1] + p0.z * h[2] + p0.w * h[3]
                 + p1.x * h[4] + p1.y * h[5] + p1.z * h[6] + p1.w * h[7];
        sG[w][gr][r] = gv;
      }
      if (lane < 2) {                        // 1/s (rcp + 1 Newton) and innov
        const int gr = lane;
        const float4 ga = *(const float4*)&sG[w][gr][0];
        const float4 gb = *(const float4*)&sG[w][gr][4];
        const float4 ma = *(const float4*)&sMu[w][gr][0];
        const float4 mb = *(const float4*)&sMu[w][gr][4];
        float s = 1.f + ga.x * h[0] + ga.y * h[1] + ga.z * h[2] + ga.w * h[3]
                      + gb.x * h[4] + gb.y * h[5] + gb.z * h[6] + gb.w * h[7];
        float hm = ma.x * h[0] + ma.y * h[1] + ma.z * h[2] + ma.w * h[3]
                 + mb.x * h[4] + mb.y * h[5] + mb.z * h[6] + mb.w * h[7];
        float inv = __builtin_amdgcn_rcpf(s);
        inv = inv * (2.f - s * inv);         // Newton refine
        sSc[w][gr][0] = inv;
        sSc[w][gr][1] = sZp[w][gr][i] - hm;
      }
      if (lane < 16) {                       // mu += g * (innov/s)
        const int gr = lane >> 3, r = lane & 7;
        const float kfac = sSc[w][gr][1] * sSc[w][gr][0];
        sMu[w][gr][r] += sG[w][gr][r] * kfac;
      }
      {                                      // P -= g g^T / s  (all 32 lanes, b128 rows)
        const int gr = lane >> 4;            // group = half
        const int l16 = lane & 15;
        const int r = l16 >> 1, cb = (l16 & 1) * 4;
        const float sc = sG[w][gr][r] * sSc[w][gr][0];   // g_r / s
        const float4 gc = *(const float4*)&sG[w][gr][cb];
        float4 p = *(const float4*)&sP[w][gr][r * 8 + cb];
        p.x -= sc * gc.x;
        p.y -= sc * gc.y;
        p.z -= sc * gc.z;
        p.w -= sc * gc.w;
        *(float4*)&sP[w][gr][r * 8 + cb] = p;
      }
    }
  }
}

extern "C" void kernel_launch(void* const* d_in, const int* in_sizes, int n_in,
                              void* d_out, int out_size, void* d_ws, size_t ws_size,
                              hipStream_t stream) {
  (void)in_sizes; (void)n_in; (void)out_size; (void)ws_size;
  const float* x  = (const float*)d_in[0];
  const float* F  = (const float*)d_in[1];
  const float* H  = (const float*)d_in[2];
  const float* Q  = (const float*)d_in[3];
  const float* R  = (const float*)d_in[4];
  const float* s0 = (const float*)d_in[5];
  float* ws = (float*)d_ws;

  kf_precompute<<<1, 1, 0, stream>>>(R, H, ws);
  // 512 blocks x 256 threads = 8 waves/block, 2 groups/wave -> 8192 groups
  kf_kernel<<<KF_B / 16, 256, 0, stream>>>(x, F, H, Q, s0, ws, (float*)d_out);
}